// CECJSLoss_55851754717292
// MI455X (gfx1250) — compile-verified
//
#include <hip/hip_runtime.h>
#include <stdint.h>

// CE + CJS loss over pred_logit[B,C], gt[B,C], C = 8192, f32.
// Memory-bound: 256 MiB / 23.3 TB/s ~= 11.5 us floor. One HBM pass per input:
// pred kept in registers, gt staged to LDS via CDNA5 async DMA overlapped with
// the softmax normalizer phases. WMMA f32 16x16x4 (ones-matrix) wave reductions.

#define BLOCK 256
#define NWAVES (BLOCK / 32)
#define C_DIM 8192
#define V4PT (C_DIM / (4 * BLOCK)) // 8 float4 per thread
#define EPSF 1e-8f

typedef float f4  __attribute__((ext_vector_type(4)));
typedef float v2f __attribute__((ext_vector_type(2)));
typedef float v8f __attribute__((ext_vector_type(8)));

// ---------------- CDNA5 async global -> LDS staging ----------------
#if defined(__gfx1250__) && __has_builtin(__builtin_amdgcn_global_load_async_to_lds_b128)
#define HAVE_ASYNC_LDS 1
// Builtin signature (from hipcc diagnostic): param0 = v4i in addrspace(1),
// param1 = v4i in addrspace(3), then imm offset, imm cpol.
typedef int v4i_g __attribute__((vector_size(16)));
typedef __attribute__((address_space(1))) v4i_g* GA4;
typedef __attribute__((address_space(3))) v4i_g* LA4;
__device__ __forceinline__ void stage16(const void* gsrc, void* ldst) {
  // 16B per lane, ASYNCcnt-tracked; LDS operand is the 32-bit wave-relative offset.
  __builtin_amdgcn_global_load_async_to_lds_b128(
      (GA4)(uintptr_t)gsrc, (LA4)(uint32_t)(uintptr_t)ldst, 0, 0);
}
__device__ __forceinline__ void stage_wait() {
#if __has_builtin(__builtin_amdgcn_s_wait_asynccnt)
  __builtin_amdgcn_s_wait_asynccnt(0);
#else
  asm volatile("s_wait_asynccnt 0" ::: "memory");
#endif
}
#else
#define HAVE_ASYNC_LDS 0
__device__ __forceinline__ void stage16(const void* gsrc, void* ldst) {
  *(f4*)ldst = *(const f4*)gsrc;
}
__device__ __forceinline__ void stage_wait() {}
#endif

// ---------------- WMMA-based wave32 sum reduction ----------------
// A(16x4) = {x, 0} per lane => A[m][0]=x_m (lanes 0-15), A[m][2]=x_{m+16}.
// B(4x16) = all ones (layout-independent). D[m][n] = x_m + x_{m+16}.
// Per-lane sum of the 8 D VGPRs covers rows {0..7} (lanes<16) / {8..15}
// (lanes>=16); one shfl_xor(16) completes the 32-lane sum, broadcast to all.
#if defined(__gfx1250__) && __has_builtin(__builtin_amdgcn_wmma_f32_16x16x4_f32)
__device__ __forceinline__ float wave_sum(float x) {
  v2f a; a[0] = x;   a[1] = 0.0f;
  v2f b; b[0] = 1.0f; b[1] = 1.0f;
  v8f c = {0.f, 0.f, 0.f, 0.f, 0.f, 0.f, 0.f, 0.f};
  c = __builtin_amdgcn_wmma_f32_16x16x4_f32(false, a, false, b, (short)0, c,
                                            false, false);
  float s = ((c[0] + c[1]) + (c[2] + c[3])) + ((c[4] + c[5]) + (c[6] + c[7]));
  s += __shfl_xor(s, 16, 32);
  return s;
}
#else
__device__ __forceinline__ float wave_sum(float x) {
#pragma unroll
  for (int off = 16; off; off >>= 1) x += __shfl_xor(x, off, 32);
  return x;
}
#endif

__device__ __forceinline__ float wave_max(float x) {
#pragma unroll
  for (int off = 16; off; off >>= 1) x = fmaxf(x, __shfl_xor(x, off, 32));
  return x;
}

// Block-wide sum, result broadcast to all threads. Trailing barrier allows
// immediate reuse of `red`.
__device__ __forceinline__ float block_sum(float x, float* red) {
  const int wv = threadIdx.x >> 5, lane = threadIdx.x & 31;
  float s = wave_sum(x);
  if (lane == 0) red[wv] = s;
  __syncthreads();
  float tot = 0.f;
#pragma unroll
  for (int w = 0; w < NWAVES; ++w) tot += red[w];
  __syncthreads();
  return tot;
}

// ---------------- Kernel 1: one block per row ----------------
__global__ __launch_bounds__(BLOCK) void cecjs_row_kernel(
    const float* __restrict__ pred, const float* __restrict__ gt,
    float* __restrict__ partial, int rows) {
  __shared__ f4 sgt[C_DIM / 4];    // 32 KB staged gt row
  __shared__ float red[NWAVES];

  const int row = blockIdx.x;
  const int tid = threadIdx.x;
  const int wv = tid >> 5, lane = tid & 31;

  const f4* predRow = (const f4*)(pred + (size_t)row * C_DIM);
  const f4* gtRow   = (const f4*)(gt   + (size_t)row * C_DIM);

  // Kick off async DMA of the gt row into LDS; overlaps the softmax phases.
#pragma unroll
  for (int k = 0; k < V4PT; ++k) {
    const int vec = k * BLOCK + tid;
    stage16(&gtRow[vec], &sgt[vec]);
  }

  // pred row -> registers (single HBM read, non-temporal: no reuse, > L2).
  f4 x[V4PT];
#pragma unroll
  for (int k = 0; k < V4PT; ++k)
    x[k] = __builtin_nontemporal_load(&predRow[k * BLOCK + tid]);

  // ---- phase 1: row max ----
  float mx = -__builtin_inff();
#pragma unroll
  for (int k = 0; k < V4PT; ++k) {
    mx = fmaxf(mx, fmaxf(fmaxf(x[k][0], x[k][1]), fmaxf(x[k][2], x[k][3])));
  }
  mx = wave_max(mx);
  if (lane == 0) red[wv] = mx;
  __syncthreads();
  float rowmax = red[0];
#pragma unroll
  for (int w = 1; w < NWAVES; ++w) rowmax = fmaxf(rowmax, red[w]);
  __syncthreads();

  // ---- phase 2: sum exp(x - max) ----
  float se = 0.f;
#pragma unroll
  for (int k = 0; k < V4PT; ++k) {
#pragma unroll
    for (int j = 0; j < 4; ++j) se += __expf(x[k][j] - rowmax);
  }
  const float logZ = rowmax + __logf(block_sum(se, red));

  // ---- phase 3: elementwise CE + weighted CJS contributions ----
  stage_wait();      // this wave's async DMA into its own sgt slots is done
  float ce_acc = 0.f, cjs_acc = 0.f;
#pragma unroll
  for (int k = 0; k < V4PT; ++k) {
    const int vec = k * BLOCK + tid;
    const f4 g = sgt[vec];
    const f4 xx = x[k];
    const float wbase = (float)(C_DIM - vec * 4);
#pragma unroll
    for (int j = 0; j < 4; ++j) {
      const float lp = xx[j] - logZ;     // log p
      const float p  = __expf(lp);
      const float gv = g[j];
      ce_acc = fmaf(gv, lp, ce_acc);
      const float lm = __logf(0.5f * (gv + p + EPSF));   // log m
      float t = p * (lp - lm);                           // xlogy(p,p) - p*logm
      t += (gv > 0.f) ? gv * (__logf(gv) - lm) : 0.f;    // xlogy(g,g) - g*logm
      cjs_acc = fmaf(t, wbase - (float)j, cjs_acc);      // * (C - c)
    }
  }

  const float ce_tot  = block_sum(ce_acc, red);
  const float cjs_tot = block_sum(cjs_acc, red);
  if (tid == 0) {
    partial[row] = ce_tot;
    partial[rows + row] = cjs_tot;
  }
}

// ---------------- Kernel 2: fold 2*rows partials into the scalar ----------------
__global__ __launch_bounds__(BLOCK) void cecjs_final_kernel(
    const float* __restrict__ partial, float* __restrict__ out, int rows) {
  __shared__ float red[NWAVES];
  float ce = 0.f, cj = 0.f;
  for (int i = threadIdx.x; i < rows; i += BLOCK) {
    ce += partial[i];
    cj += partial[rows + i];
  }
  const float ce_t = block_sum(ce, red);
  const float cj_t = block_sum(cj, red);
  if (threadIdx.x == 0) {
    // loss = -ce_sum/B + 0.5 * (0.5 * cjs_sum / B)
    out[0] = (0.25f * cj_t - ce_t) / (float)rows;
  }
}

extern "C" void kernel_launch(void* const* d_in, const int* in_sizes, int n_in,
                              void* d_out, int out_size, void* d_ws, size_t ws_size,
                              hipStream_t stream) {
  const float* pred = (const float*)d_in[0];
  const float* gt   = (const float*)d_in[1];
  float* ws = (float*)d_ws;             // needs 2 * rows * 4 bytes (32 KB)
  float* out = (float*)d_out;
  const int rows = in_sizes[0] / C_DIM; // 4096

  hipLaunchKernelGGL(cecjs_row_kernel, dim3(rows), dim3(BLOCK), 0, stream,
                     pred, gt, ws, rows);
  hipLaunchKernelGGL(cecjs_final_kernel, dim3(1), dim3(BLOCK), 0, stream,
                     ws, out, rows);
}